// CenterLoss_90142773609056
// MI455X (gfx1250) — compile-verified
//
#include <hip/hip_runtime.h>

typedef __attribute__((ext_vector_type(2))) float v2f;
typedef __attribute__((ext_vector_type(8))) float v8f;

#define D_DIM 512
#define B_ROWS 4096
#define C_CLASSES 10000

// One wave (32 lanes) handles 16 batch rows via V_WMMA_F32_16X16X4_F32.
// A-tile (16x4 of x) and B-tile (4x16 of gathered centers) share the same
// per-lane addressing pattern:
//   lane L (0..15):  elements k+0..1 of row L
//   lane L+16:       elements k+2..3 of row L
// Accumulate D = A*B over K=512 (128 wmma steps); diagonal D[m,m] = x_m . c_lab(m).
// ||x||^2, ||c||^2 accumulate in VALU, combined across the half-waves by shfl_xor(16).
__global__ __launch_bounds__(256) void center_loss_rows_kernel(
    const float* __restrict__ x,        // [B_ROWS, D_DIM]
    const float* __restrict__ centers,  // [C_CLASSES, D_DIM]
    const int* __restrict__ labels,     // [B_ROWS] (int32; JAX x64 disabled)
    float* __restrict__ partials)       // [gridDim.x]
{
    const int lane = threadIdx.x & 31;
    const int wave = threadIdx.x >> 5;
    const int rowbase = (blockIdx.x * 8 + wave) * 16;
    const int r = rowbase + (lane & 15);
    const int koff = (lane >> 4) << 1;  // 0 for lanes 0-15, 2 for lanes 16-31

    const float* __restrict__ xrow = x + (size_t)r * D_DIM + koff;
    const int lab = labels[r];
    const float* __restrict__ crow = centers + (size_t)lab * D_DIM + koff;

    v8f acc = {0.f, 0.f, 0.f, 0.f, 0.f, 0.f, 0.f, 0.f};
    float xx = 0.f, cc = 0.f;

#pragma unroll 8
    for (int k = 0; k < D_DIM; k += 4) {
        v2f a = *(const v2f*)(xrow + k);  // x[r, k+koff .. k+koff+1]
        v2f b = *(const v2f*)(crow + k);  // centers[lab, k+koff .. k+koff+1]
        // D = A x B + C   (16x16x4 f32)
        acc = __builtin_amdgcn_wmma_f32_16x16x4_f32(
            /*neg_a=*/false, a, /*neg_b=*/false, b,
            /*c_mod=*/(short)0, acc, /*reuse_a=*/false, /*reuse_b=*/false);
        xx = fmaf(a.x, a.x, fmaf(a.y, a.y, xx));
        cc = fmaf(b.x, b.x, fmaf(b.y, b.y, cc));
    }

    // Row totals: lane L and lane L+16 each saw half of the K dimension.
    xx += __shfl_xor(xx, 16, 32);
    cc += __shfl_xor(cc, 16, 32);

    // Diagonal extraction per C/D layout:
    //   m = 0..7  -> VGPR m,   lane m       (N = lane)
    //   m = 8..15 -> VGPR m-8, lane m+16    (N = lane-16)
    const int di = (lane < 8) ? lane : (lane - 24);  // valid 0..7 only on diag lanes
    float xc = 0.f;
#pragma unroll
    for (int j = 0; j < 8; ++j) xc = (di == j) ? acc[j] : xc;

    float val = 0.f;
    if ((lane < 8) || (lane >= 24)) {
        float dm = 0.5f * (xx + cc) + 0.3f * xc;
        dm = fminf(fmaxf(dm, 1e-12f), 1e12f);  // clip, faithful to reference
        val = dm;
    }

    // Wave reduction (fixed tree -> deterministic)
#pragma unroll
    for (int off = 16; off >= 1; off >>= 1) val += __shfl_xor(val, off, 32);

    __shared__ float wsum[8];
    if (lane == 0) wsum[wave] = val;
    __syncthreads();
    if (threadIdx.x == 0) {
        float s = 0.f;
#pragma unroll
        for (int w = 0; w < 8; ++w) s += wsum[w];
        partials[blockIdx.x] = s;
    }
}

__global__ __launch_bounds__(32) void center_loss_finalize_kernel(
    const float* __restrict__ partials, float* __restrict__ out, int n)
{
    float v = ((int)threadIdx.x < n) ? partials[threadIdx.x] : 0.f;
#pragma unroll
    for (int off = 16; off >= 1; off >>= 1) v += __shfl_xor(v, off, 32);
    if (threadIdx.x == 0) {
        // + (C-1) masked entries per row, each clipped up to 1e-12
        out[0] = v * (1.0f / (float)B_ROWS) + (float)(C_CLASSES - 1) * 1e-12f;
    }
}

extern "C" void kernel_launch(void* const* d_in, const int* in_sizes, int n_in,
                              void* d_out, int out_size, void* d_ws, size_t ws_size,
                              hipStream_t stream) {
    const float* x       = (const float*)d_in[0];  // [4096, 512]
    const float* centers = (const float*)d_in[1];  // [10000, 512]
    const int*   labels  = (const int*)d_in[2];    // [4096]
    float* out = (float*)d_out;
    float* partials = (float*)d_ws;  // 32 floats, fully overwritten every call

    const int nblocks = B_ROWS / 128;  // 128 rows per 256-thread block -> 32
    center_loss_rows_kernel<<<nblocks, 256, 0, stream>>>(x, centers, labels, partials);
    center_loss_finalize_kernel<<<1, 32, 0, stream>>>(partials, out, nblocks);
}